// VGAE_encoder_77867757077116
// MI455X (gfx1250) — compile-verified
//
#include <hip/hip_runtime.h>

typedef __attribute__((ext_vector_type(2))) float v2f;
typedef __attribute__((ext_vector_type(8))) float v8f;

// ---------------------------------------------------------------------------
// 1) Degree / normalization: deg starts at 1 (self loop), +1 per incoming edge,
//    then dis = rsqrt(deg) in place.  (deg >= 1 always, matching reference.)
// ---------------------------------------------------------------------------
__global__ void init_deg_kernel(float* __restrict__ deg, int n) {
  int t = blockIdx.x * blockDim.x + threadIdx.x;
  if (t < n) deg[t] = 1.0f;
}

__global__ void degree_kernel(const int* __restrict__ dst, float* __restrict__ deg, int E) {
  int t = blockIdx.x * blockDim.x + threadIdx.x;
  if (t < E) atomicAdd(&deg[dst[t]], 1.0f);
}

__global__ void rsqrt_kernel(float* __restrict__ deg, int n) {
  int t = blockIdx.x * blockDim.x + threadIdx.x;
  if (t < n) deg[t] = rsqrtf(deg[t]);
}

// ---------------------------------------------------------------------------
// 2) Dense transform: W staged in LDS once per block; each wave computes a
//    16 x N output strip with N/16 independent WMMA accumulators, so each A
//    fragment is loaded once per K-step and reused across all N-tiles.
//
//    V_WMMA_F32_16X16X4_F32 fragment layout (ISA 7.12.2):
//      A 16x4:  lanes 0-15 -> M=lane, VGPR0/1 = K0/K1 ; lanes 16-31 -> K2/K3
//      B 4x16:  lanes 0-15 -> N=lane, VGPR0/1 = K0/K1 ; lanes 16-31 -> K2/K3
//      C 16x16: VGPR v, lane l: M = v + (l>=16 ? 8 : 0), N = l&15
//
//    LDS row stride = N+8 floats: the two half-wave rows sit 2*(N+8) = 16
//    (mod 64) banks apart, so their 16-lane bank ranges never overlap.
//
//    Epilogue scales row m by dis[m] and writes BOTH g (gather source) and
//    agg (accumulator pre-seeded with the self-loop contribution g[m]).
// ---------------------------------------------------------------------------
template <int K, int N>
__global__ void __launch_bounds__(128)
gemm_scale_kernel(const float* __restrict__ X,   // [M,K] row-major
                  const float* __restrict__ W,   // [K,N] row-major
                  const float* __restrict__ dis, // [M]
                  float* __restrict__ g,         // [M,N]
                  float* __restrict__ agg,       // [M,N]
                  int M) {
  constexpr int NT = N / 16;          // N-tiles per wave (8 or 4)
  constexpr int STRIDE = N + 8;       // padded LDS row stride (floats)
  __shared__ float sW[K * STRIDE];

  // Cooperative stage of W into LDS (float4 global load -> ds_store_b128).
  constexpr int C4 = N / 4;
  for (int i = threadIdx.x; i < K * C4; i += blockDim.x) {
    const int row = i / C4;
    const int c4  = (i - row * C4) * 4;
    const float4 w4 = *reinterpret_cast<const float4*>(W + (size_t)row * N + c4);
    float* p = &sW[row * STRIDE + c4];
    p[0] = w4.x; p[1] = w4.y; p[2] = w4.z; p[3] = w4.w;
  }
  __syncthreads();

  const int wave = blockIdx.x * (blockDim.x >> 5) + (threadIdx.x >> 5);
  if (wave >= (M >> 4)) return;       // uniform per wave; after the barrier
  const int m0   = wave << 4;
  const int lane = threadIdx.x & 31;
  const int half = lane >> 4;         // 0: K pair 0/1, 1: K pair 2/3
  const int l16  = lane & 15;

  const float* __restrict__ arow = X + (size_t)(m0 + l16) * K;

  v8f acc[NT];
#pragma unroll
  for (int t = 0; t < NT; ++t) acc[t] = (v8f){};

#pragma unroll 2
  for (int k0 = 0; k0 < K; k0 += 4) {
    const int ka = k0 + (half << 1);
    v2f a;
    a.x = arow[ka];
    a.y = arow[ka + 1];
    const float* __restrict__ brow0 = &sW[ka * STRIDE];
    const float* __restrict__ brow1 = &sW[(ka + 1) * STRIDE];
#pragma unroll
    for (int t = 0; t < NT; ++t) {
      v2f b;
      b.x = brow0[t * 16 + l16];
      b.y = brow1[t * 16 + l16];
      acc[t] = __builtin_amdgcn_wmma_f32_16x16x4_f32(
          /*neg_a=*/false, a, /*neg_b=*/false, b,
          /*c_mod=*/(short)0, acc[t], /*reuse_a=*/false, /*reuse_b=*/false);
    }
  }

#pragma unroll
  for (int t = 0; t < NT; ++t) {
#pragma unroll
    for (int v = 0; v < 8; ++v) {
      const int mr = m0 + v + (half << 3);
      const float val = dis[mr] * acc[t][v];
      const size_t idx = (size_t)mr * N + t * 16 + l16;
      g[idx]   = val;   // gather source for edge scatter
      agg[idx] = val;   // self-loop contribution seeds the accumulator
    }
  }
}

// ---------------------------------------------------------------------------
// 3) Edge scatter: agg[dst] += g[src].  One thread per (edge, 4-channel chunk):
//    float4 coalesced gather (L2-resident) + 4 global_atomic_add_f32.
// ---------------------------------------------------------------------------
template <int CHUNKS>
__global__ void scatter_kernel(const int* __restrict__ src, const int* __restrict__ dst,
                               const float* __restrict__ g, float* __restrict__ agg, int E) {
  const int C = CHUNKS * 4;
  int t = blockIdx.x * blockDim.x + threadIdx.x;
  if (t >= E * CHUNKS) return;
  const int e  = t / CHUNKS;
  const int c4 = (t - e * CHUNKS) << 2;
  const int s = src[e];
  const int d = dst[e];
  const float4 v = *reinterpret_cast<const float4*>(g + (size_t)s * C + c4);
  float* ap = agg + (size_t)d * C + c4;
  atomicAdd(ap + 0, v.x);
  atomicAdd(ap + 1, v.y);
  atomicAdd(ap + 2, v.z);
  atomicAdd(ap + 3, v.w);
}

// ---------------------------------------------------------------------------
// 4) Epilogues: out = dis[row]*agg + b  (optionally ReLU), in place or to d_out.
// ---------------------------------------------------------------------------
template <int C>
__global__ void bias_relu_kernel(float* __restrict__ a, const float* __restrict__ dis,
                                 const float* __restrict__ b, int total) {
  int t = blockIdx.x * blockDim.x + threadIdx.x;
  if (t >= total) return;
  const int i = t / C;
  const int c = t - i * C;
  const float v = dis[i] * a[t] + b[c];
  a[t] = v > 0.0f ? v : 0.0f;
}

template <int C>
__global__ void bias_out_kernel(const float* __restrict__ a, const float* __restrict__ dis,
                                const float* __restrict__ b, float* __restrict__ o, int total) {
  int t = blockIdx.x * blockDim.x + threadIdx.x;
  if (t >= total) return;
  const int i = t / C;
  const int c = t - i * C;
  o[t] = dis[i] * a[t] + b[c];
}

// ---------------------------------------------------------------------------
// Launcher
// ---------------------------------------------------------------------------
extern "C" void kernel_launch(void* const* d_in, const int* in_sizes, int n_in,
                              void* d_out, int out_size, void* d_ws, size_t ws_size,
                              hipStream_t stream) {
  const float* x  = (const float*)d_in[0];
  const int*   ei = (const int*)d_in[1];
  const float* W1 = (const float*)d_in[2];
  const float* b1 = (const float*)d_in[3];
  const float* W2 = (const float*)d_in[4];
  const float* b2 = (const float*)d_in[5];
  float* out = (float*)d_out;

  const int HID = in_sizes[3];            // 128
  const int OUT = in_sizes[5];            // 64
  const int IN  = in_sizes[2] / HID;      // 128
  const int M   = in_sizes[0] / IN;       // 50000 (multiple of 16)
  const int E   = in_sizes[1] / 2;        // 800000
  const int* src = ei;                    // edge_index row 0
  const int* dst = ei + E;                // edge_index row 1

  // Workspace layout (floats): dis | g1 | agg1(=a1 in place) | g2 | agg2
  float* ws = (float*)d_ws;
  size_t off = 0;
  float* dis  = ws + off; off += (size_t)((M + 63) & ~63);
  float* g1   = ws + off; off += (size_t)M * HID;
  float* agg1 = ws + off; off += (size_t)M * HID;
  float* g2   = ws + off; off += (size_t)M * OUT;
  float* agg2 = ws + off;

  const int TB = 256;

  // --- normalization ---
  init_deg_kernel<<<(M + TB - 1) / TB, TB, 0, stream>>>(dis, M);
  degree_kernel<<<(E + TB - 1) / TB, TB, 0, stream>>>(dst, dis, E);
  rsqrt_kernel<<<(M + TB - 1) / TB, TB, 0, stream>>>(dis, M);

  // --- layer 1: g1 = dis ⊙ (x @ W1); agg1 seeded with g1 (self loop) ---
  {
    const int strips = M / 16;                       // 16-row strips
    gemm_scale_kernel<128, 128><<<(strips + 3) / 4, 128, 0, stream>>>(x, W1, dis, g1, agg1, M);
  }
  {
    const long long tt = (long long)E * (HID / 4);
    scatter_kernel<32><<<(int)((tt + TB - 1) / TB), TB, 0, stream>>>(src, dst, g1, agg1, E);
  }
  {
    const int total = M * HID;   // a1 = relu(dis*agg1 + b1), in place
    bias_relu_kernel<128><<<(total + TB - 1) / TB, TB, 0, stream>>>(agg1, dis, b1, total);
  }

  // --- layer 2: g2 = dis ⊙ (a1 @ W2); agg2 seeded with g2 ---
  {
    const int strips = M / 16;
    gemm_scale_kernel<128, 64><<<(strips + 3) / 4, 128, 0, stream>>>(agg1, W2, dis, g2, agg2, M);
  }
  {
    const long long tt = (long long)E * (OUT / 4);
    scatter_kernel<16><<<(int)((tt + TB - 1) / TB), TB, 0, stream>>>(src, dst, g2, agg2, E);
  }
  {
    const int total = M * OUT;   // out = dis*agg2 + b2
    bias_out_kernel<64><<<(total + TB - 1) / TB, TB, 0, stream>>>(agg2, dis, b2, out, total);
  }
}